// RX_2448131359093
// MI455X (gfx1250) — compile-verified
//
#include <hip/hip_runtime.h>
#include <math.h>

typedef __attribute__((ext_vector_type(2))) float v2f;
typedef __attribute__((ext_vector_type(8))) float v8f;

namespace {
constexpr int kNQ      = 14;
constexpr int kState   = 1 << kNQ;          // 16384
constexpr int kInner   = 16;                // H16 axis (low 4 bits)
constexpr int kOuter   = kState / kInner;   // 1024  (high 10 bits)
constexpr int kPad     = 18;                // padded dwords per outer row (bank-conflict free, 8B aligned)
constexpr int kThreads = 512;               // 16 waves of 32
constexpr int kPhHiOff = 2 * kOuter * kPad; // dword offset of ph_hi table
constexpr int kPhLoOff = kPhHiOff + 128;
constexpr int kLdsDwords = kPhLoOff + 128;  // 37120 dwords = 148480 bytes
}

// H16[r][c] = (-1)^popcount(r&c)  (unnormalized Walsh-Hadamard, symmetric)
__device__ __forceinline__ float hadsign(int r, int c) {
  return (__popc(r & c) & 1) ? -1.0f : 1.0f;
}

__global__ __launch_bounds__(kThreads)
void rx_fwht_kernel(const float* __restrict__ phi_real,
                    const float* __restrict__ phi_imag,
                    const float* __restrict__ thetas,
                    float* __restrict__ out) {
  extern __shared__ float lds[];
  float* xr    = lds;
  float* xi    = lds + kOuter * kPad;
  float* ph_hi = lds + kPhHiOff;
  float* ph_lo = lds + kPhLoOff;

  const int tid  = threadIdx.x;
  const int row  = blockIdx.x;
  const int lane = tid & 31;
  const int wave = tid >> 5;
  const int n    = lane & 15;   // N index / M index inside wave
  const int d    = lane >> 4;   // lane-half selects K pair (WMMA f32 layout)

  // ---------- load row into padded LDS layout: element (o,i) -> dword o*18+i ----------
  const float* pr = phi_real + (size_t)row * kState;
  const float* pim = phi_imag + (size_t)row * kState;
  #pragma unroll
  for (int r = 0; r < kState / 4 / kThreads; ++r) {   // 8 float4 per thread
    int f4 = tid + kThreads * r;
    int e  = f4 << 2;                                 // element index, multiple of 4
    float4 vr = reinterpret_cast<const float4*>(pr)[f4];
    float4 vi = reinterpret_cast<const float4*>(pim)[f4];
    int idx = (e >> 4) * kPad + (e & 15);
    *reinterpret_cast<float2*>(&xr[idx])     = make_float2(vr.x, vr.y);
    *reinterpret_cast<float2*>(&xr[idx + 2]) = make_float2(vr.z, vr.w);
    *reinterpret_cast<float2*>(&xi[idx])     = make_float2(vi.x, vi.y);
    *reinterpret_cast<float2*>(&xi[idx + 2]) = make_float2(vi.z, vi.w);
  }

  // ---------- separable phase tables: phase(s) = ph_hi[s>>7] + ph_lo[s&127] ----------
  if (tid < 256) {
    const float* th = thetas + row * kNQ;
    int which = tid >> 7;      // 0 -> hi (qubits 0..6), 1 -> lo (qubits 7..13)
    int v = tid & 127;
    float acc = 0.0f;
    #pragma unroll
    for (int q = 0; q < 7; ++q) {
      float t = th[(which ? 7 : 0) + q];
      acc += ((v >> (6 - q)) & 1) ? t : 0.0f;
    }
    (which ? ph_lo : ph_hi)[v] = acc;
  }

  // ---------- constant H16 operand (B matrix) registers ----------
  // B[k][n] layout: lane = n + 16*(k>>1), vgpr = k&1; slice ks covers k = 4ks..4ks+3.
  v2f b0 = { hadsign(0  + 2*d, n), hadsign(1  + 2*d, n) };
  v2f b1 = { hadsign(4  + 2*d, n), hadsign(5  + 2*d, n) };
  v2f b2 = { hadsign(8  + 2*d, n), hadsign(9  + 2*d, n) };
  v2f b3 = { hadsign(12 + 2*d, n), hadsign(13 + 2*d, n) };

  __syncthreads();

  for (int pass = 0; pass < 2; ++pass) {
    // ===== inner H16 via WMMA: D(16 o-rows x 16 i') = A(data 16x16) * B(H16), K chained 4x4 =====
    #pragma unroll
    for (int it = 0; it < 8; ++it) {
      int job = wave * 8 + it;          // 128 jobs = 64 tiles x {re,im}
      float* x = (job & 1) ? xi : xr;
      int o0 = (job >> 1) << 4;
      // A[m][k]: lane = m + 16*(k>>1), vgpr = k&1  -> consecutive dword pair per slice
      int rowbase = (o0 + n) * kPad + 2 * d;
      v2f a0 = *reinterpret_cast<v2f*>(&x[rowbase + 0]);
      v2f a1 = *reinterpret_cast<v2f*>(&x[rowbase + 4]);
      v2f a2 = *reinterpret_cast<v2f*>(&x[rowbase + 8]);
      v2f a3 = *reinterpret_cast<v2f*>(&x[rowbase + 12]);
      v8f c = {};
      c = __builtin_amdgcn_wmma_f32_16x16x4_f32(false, a0, false, b0, (short)0, c, false, false);
      c = __builtin_amdgcn_wmma_f32_16x16x4_f32(false, a1, false, b1, (short)0, c, false, false);
      c = __builtin_amdgcn_wmma_f32_16x16x4_f32(false, a2, false, b2, (short)0, c, false, false);
      c = __builtin_amdgcn_wmma_f32_16x16x4_f32(false, a3, false, b3, (short)0, c, false, false);
      // D[m][n']: vgpr v -> rows v (lanes 0-15) / v+8 (lanes 16-31), col = lane&15
      #pragma unroll
      for (int v = 0; v < 8; ++v)
        x[(o0 + v + 8 * d) * kPad + n] = c[v];
    }
    __syncthreads();

    // ===== H1024 on outer axis: 10 radix-2 butterfly stages in LDS =====
    for (int lh = 0; lh < 10; ++lh) {
      int h = 1 << lh;
      #pragma unroll
      for (int r = 0; r < 16; ++r) {
        int p  = tid + kThreads * r;          // 0..8191 butterflies
        int i  = p & 15;
        int j  = p >> 4;                      // 0..511 outer pair id
        int lo = j & (h - 1);
        int oa = ((j >> lh) << (lh + 1)) + lo;
        int ia = oa * kPad + i;
        int ib = ia + h * kPad;
        float ar = xr[ia], br = xr[ib];
        float ai = xi[ia], bi = xi[ib];
        xr[ia] = ar + br; xr[ib] = ar - br;
        xi[ia] = ai + bi; xi[ib] = ai - bi;
      }
      __syncthreads();
    }

    if (pass == 0) {
      // ===== multiply by exp(-0.5i * phase): (a+ib)(c - is) =====
      #pragma unroll
      for (int r = 0; r < 32; ++r) {
        int p   = tid + kThreads * r;         // s = p
        int idx = (p >> 4) * kPad + (p & 15);
        float ph = 0.5f * (ph_hi[p >> 7] + ph_lo[p & 127]);
        float sn, cs;
        sincosf(ph, &sn, &cs);
        float a = xr[idx], b = xi[idx];
        xr[idx] = a * cs + b * sn;
        xi[idx] = b * cs - a * sn;
      }
      __syncthreads();
    }
  }

  // ---------- writeout: interleaved complex64, total scale 2^-14 ----------
  const float scale = 1.0f / 16384.0f;
  float2* out2 = reinterpret_cast<float2*>(out) + (size_t)row * kState;
  #pragma unroll
  for (int r = 0; r < 32; ++r) {
    int p   = tid + kThreads * r;
    int idx = (p >> 4) * kPad + (p & 15);
    out2[p] = make_float2(xr[idx] * scale, xi[idx] * scale);
  }
}

extern "C" void kernel_launch(void* const* d_in, const int* in_sizes, int n_in,
                              void* d_out, int out_size, void* d_ws, size_t ws_size,
                              hipStream_t stream) {
  const float* phi_real = (const float*)d_in[0];
  const float* phi_imag = (const float*)d_in[1];
  const float* thetas   = (const float*)d_in[2];
  float* out = (float*)d_out;
  int batch = in_sizes[2] / kNQ;              // 512
  size_t lds_bytes = (size_t)kLdsDwords * sizeof(float);
  rx_fwht_kernel<<<dim3(batch), dim3(kThreads), lds_bytes, stream>>>(
      phi_real, phi_imag, thetas, out);
}